// Clements_Arct_32641751450013
// MI455X (gfx1250) — compile-verified
//
#include <hip/hip_runtime.h>
#include <math.h>

#define NN      512
#define PE      256
#define PO      255
#define NLAY_E  512            // even layers (odd layer count identical)
#define LAYERS  1024           // 4 sub-layers per step * 256 steps
#define NSEG    4
#define SEG_LAYERS (LAYERS / NSEG)   // 256
#define CSLAB   8              // columns per workgroup (chain kernel)
#define TPB     256
#define TWO_PI_F 6.283185307179586f

// ---- CDNA5 async global->LDS path (probe-verified signature) ----
#if defined(__AMDGCN__) && __has_builtin(__builtin_amdgcn_global_load_async_to_lds_b128)
#define USE_ASYNC_LDS 1
#else
#define USE_ASYNC_LDS 0
#endif
#if defined(__AMDGCN__) && __has_builtin(__builtin_amdgcn_s_wait_asynccnt)
#define HAVE_WAIT_ASYNC 1
#else
#define HAVE_WAIT_ASYNC 0
#endif

typedef int   v4i_t __attribute__((vector_size(16)));
typedef float v2f   __attribute__((ext_vector_type(2)));
typedef float v8f   __attribute__((ext_vector_type(8)));

__device__ __forceinline__ void cp16_g2lds(const float4* g, float4* l) {
#if USE_ASYNC_LDS
  __builtin_amdgcn_global_load_async_to_lds_b128(
      (__attribute__((address_space(1))) v4i_t*)g,
      (__attribute__((address_space(3))) v4i_t*)l, 0, 0);
#else
  *l = *g;
#endif
}
__device__ __forceinline__ void wait_async0() {
#if HAVE_WAIT_ASYNC
  __builtin_amdgcn_s_wait_asynccnt(0);
#elif USE_ASYNC_LDS
  asm volatile("s_wait_asynccnt 0" ::: "memory");
#endif
}

// --------------------------------------------------------------------------
// Kernel 1: per-layer coefficients (cos, sin, t, r) as float4.
// --------------------------------------------------------------------------
__global__ __launch_bounds__(TPB) void pre_k(
    const float* __restrict__ pe, const float* __restrict__ po,
    const float* __restrict__ le, const float* __restrict__ ie,
    const float* __restrict__ lo, const float* __restrict__ io,
    float4* __restrict__ evenP, float4* __restrict__ oddP) {
  int idx = blockIdx.x * TPB + threadIdx.x;
  const int half = NLAY_E * PE;
  if (idx >= 2 * half) return;
  float th, loss, imb;
  float4* dst;
  if (idx < half) {
    th = pe[idx]; loss = le[idx]; imb = ie[idx];
    dst = evenP + idx;
  } else {
    int j = idx - half;
    int k = j >> 8, p = j & 255;
    dst = oddP + j;
    if (p < PO) { int s = k * PO + p; th = po[s]; loss = lo[s]; imb = io[s]; }
    else { *dst = make_float4(1.f, 0.f, 1.f, 0.f); return; }  // padded slot
  }
  th = fminf(fmaxf(th, 0.0f), TWO_PI_F);
  float s, c;
  sincosf(th, &s, &c);
  float a = sqrtf(1.0f - loss);
  float t = a * sqrtf(0.5f + imb);
  float r = a * sqrtf(0.5f - imb);
  *dst = make_float4(c, s, t, r);
}

__device__ __forceinline__ const float4* layer_params(const float4* evenP,
                                                      const float4* oddP,
                                                      int l) {
  int s = l >> 2, q = l & 3;
  return (q < 2) ? (evenP + (size_t)(2 * s + q) * PE)
                 : (oddP  + (size_t)(2 * s + q - 2) * PE);
}

// --------------------------------------------------------------------------
// Kernel 2: segment transfer matrices. blockIdx.y = segment (256 layers),
// blockIdx.x = 8-column slab; slab LDS-resident; coefficients streamed via
// double-buffered async global->LDS (ASYNCcnt).
// --------------------------------------------------------------------------
__global__ __launch_bounds__(TPB) void chain_k(
    const float4* __restrict__ evenP, const float4* __restrict__ oddP,
    float2* __restrict__ Tseg) {
  __shared__ float2 M[NN * CSLAB];    // 32 KB slab
  __shared__ float4 pb[2][PE];        // 8 KB double-buffered coefficients

  const int tid  = threadIdx.x;
  const int col0 = blockIdx.x * CSLAB;
  const int seg  = blockIdx.y;
  const int L0   = seg * SEG_LAYERS;
  const int c    = tid & (CSLAB - 1);
  const int p0   = tid >> 3;

  for (int i = tid; i < NN * CSLAB; i += TPB) {
    int r = i >> 3, cc = i & 7;
    M[i] = make_float2((r == col0 + cc) ? 1.0f : 0.0f, 0.0f);
  }
  cp16_g2lds(layer_params(evenP, oddP, L0) + tid, &pb[0][tid]);
  wait_async0();
  __syncthreads();

  for (int l = L0; l < L0 + SEG_LAYERS; ++l) {
    const int cur = l & 1;                  // L0 is even -> starts at buffer 0
    if (l + 1 < L0 + SEG_LAYERS)
      cp16_g2lds(layer_params(evenP, oddP, l + 1) + tid, &pb[cur ^ 1][tid]);
    if (l + 8 < L0 + SEG_LAYERS)
      __builtin_prefetch((const void*)(layer_params(evenP, oddP, l + 8) + tid), 0, 0);

    const int off = (l & 3) >> 1;           // 0,0,1,1 -> even,even,odd,odd
#pragma unroll
    for (int i = 0; i < 8; ++i) {
      int p = p0 + 32 * i;
      if (!(off && p == PO)) {              // odd layers: 255 pairs only
        float4 P = pb[cur][p];              // (cos, sin, t, r)
        int e = (2 * p + off) * CSLAB + c;
        float2 top = M[e];
        float2 bot = M[e + CSLAB];
        float tr = top.x * P.x - top.y * P.y;   // phase on top row
        float ti = top.x * P.y + top.y * P.x;
        float2 nt, nb;                          // 2x2 MMI mix
        nt.x = P.z * tr    - P.w * bot.y;
        nt.y = P.z * ti    + P.w * bot.x;
        nb.x = P.z * bot.x - P.w * ti;
        nb.y = P.z * bot.y + P.w * tr;
        M[e]         = nt;
        M[e + CSLAB] = nb;
      }
    }
    wait_async0();
    __syncthreads();
  }

  float2* T = Tseg + (size_t)seg * NN * NN;
  for (int i = tid; i < NN * CSLAB; i += TPB) {
    int r = i >> 3, cc = i & 7;
    T[(size_t)r * NN + col0 + cc] = M[i];
  }
}

// --------------------------------------------------------------------------
// Kernel 3: complex GEMM C = A @ B (512x512 each) on V_WMMA_F32_16X16X4_F32.
// One wave per 16x16 output tile; 8 waves per block. Optional fused per-row
// output phase (final composition). Batched via blockIdx.z with strides.
// --------------------------------------------------------------------------
__device__ __forceinline__ v8f wmma4(v2f a, v2f b, v8f c) {
  return __builtin_amdgcn_wmma_f32_16x16x4_f32(
      false, a, false, b, (short)0, c, false, false);
}

__global__ __launch_bounds__(TPB) void gemm_k(
    const float2* __restrict__ A0, const float2* __restrict__ B0,
    float2* __restrict__ C0, long long sA, long long sB, long long sC,
    const float* __restrict__ rowPhase) {
  const float2* A = A0 + sA * blockIdx.z;
  const float2* B = B0 + sB * blockIdx.z;
  float2*       C = C0 + sC * blockIdx.z;

  const int lane = threadIdx.x & 31;
  const int wave = threadIdx.x >> 5;
  const int tile = blockIdx.x * 8 + wave;        // 0..1023 (32x32 tiles)
  const int r0 = (tile >> 5) * 16;
  const int c0 = (tile & 31) * 16;
  const int m = lane & 15;                       // row (A) / col (B,D)
  const int h = lane >> 4;                       // lane half

  v8f Dr = {0.f, 0.f, 0.f, 0.f, 0.f, 0.f, 0.f, 0.f};
  v8f Di = {0.f, 0.f, 0.f, 0.f, 0.f, 0.f, 0.f, 0.f};

  for (int k = 0; k < NN; k += 4) {
    // A operand (16x4): lane(m,h) -> A[m][k+2h], A[m][k+2h+1]  (16B aligned)
    const float4 a01 = *(const float4*)&A[(size_t)(r0 + m) * NN + k + 2 * h];
    v2f Ar = {a01.x, a01.z};
    v2f Ai = {a01.y, a01.w};
    // B operand (4x16): lane(n=m,h) -> B[k+2h][n], B[k+2h+1][n]
    const float2 b0 = B[(size_t)(k + 2 * h)     * NN + c0 + m];
    const float2 b1 = B[(size_t)(k + 2 * h + 1) * NN + c0 + m];
    v2f Br = {b0.x, b1.x};
    v2f Bi = {b0.y, b1.y};
    v2f An = {-Ai.x, -Ai.y};                     // f32 WMMA has no A-neg bit
    Dr = wmma4(Ar, Br, Dr);                      // Dr += Ar*Br
    Dr = wmma4(An, Bi, Dr);                      // Dr -= Ai*Bi
    Di = wmma4(Ar, Bi, Di);                      // Di += Ar*Bi
    Di = wmma4(Ai, Br, Di);                      // Di += Ai*Br
  }

  // D layout: VGPR j, lane(n,h) = C[r0 + j + 8h][c0 + n]
#pragma unroll
  for (int j = 0; j < 8; ++j) {
    int row = r0 + j + 8 * h;
    float re = Dr[j], im = Di[j];
    if (rowPhase) {
      float th = fminf(fmaxf(rowPhase[row], 0.0f), TWO_PI_F);
      float s, cph;
      sincosf(th, &s, &cph);
      float nre = re * cph - im * s;
      im = re * s + im * cph;
      re = nre;
    }
    C[(size_t)row * NN + c0 + m] = make_float2(re, im);
  }
}

// --------------------------------------------------------------------------
extern "C" void kernel_launch(void* const* d_in, const int* in_sizes, int n_in,
                              void* d_out, int out_size, void* d_ws, size_t ws_size,
                              hipStream_t stream) {
  (void)in_sizes; (void)n_in; (void)out_size; (void)ws_size;
  const float* pe   = (const float*)d_in[0];
  const float* po   = (const float*)d_in[1];
  const float* pout = (const float*)d_in[2];
  const float* le   = (const float*)d_in[3];
  const float* ie   = (const float*)d_in[4];
  const float* lo   = (const float*)d_in[5];
  const float* io   = (const float*)d_in[6];

  char* w = (char*)d_ws;
  float4* evenP = (float4*)w;                              // 2 MB
  float4* oddP  = (float4*)(w + ((size_t)2 << 20));        // 2 MB
  float2* Tseg  = (float2*)(w + ((size_t)4 << 20));        // 4 x 2 MB
  float2* L1    = (float2*)(w + ((size_t)12 << 20));       // 2 x 2 MB
  const long long SEGE = (long long)NN * NN;               // float2 per matrix

  pre_k<<<(2 * NLAY_E * PE + TPB - 1) / TPB, TPB, 0, stream>>>(
      pe, po, le, ie, lo, io, evenP, oddP);

  chain_k<<<dim3(NN / CSLAB, NSEG), TPB, 0, stream>>>(evenP, oddP, Tseg);

  // Level 1: z=0: L1[0] = T1 @ T0 ; z=1: L1[1] = T3 @ T2
  gemm_k<<<dim3(128, 1, 2), TPB, 0, stream>>>(
      Tseg + SEGE, Tseg, L1, 2 * SEGE, 2 * SEGE, SEGE, nullptr);

  // Final: out = phase(pc_out) * (L1[1] @ L1[0])
  gemm_k<<<dim3(128, 1, 1), TPB, 0, stream>>>(
      L1 + SEGE, L1, (float2*)d_out, 0, 0, 0, pout);
}